// InteractionDecomposition_11235634446892
// MI455X (gfx1250) — compile-verified
//
#include <hip/hip_runtime.h>
#include <hip/hip_bf16.h>

typedef __attribute__((ext_vector_type(2))) float v2f;
typedef __attribute__((ext_vector_type(8))) float v8f;

#define BSZ 64
#define D_DIM 32
#define H_DIM 256
#define NPAIR 992              // D*D - D
#define KDIM 1056              // 2*D + NPAIR
#define RIDGE 1.0e-3f

static __device__ __forceinline__ v8f wmma_f32_k4(v2f a, v2f b, v8f c) {
  // D = A(16x4,f32) * B(4x16,f32) + C ; emits v_wmma_f32_16x16x4_f32
  return __builtin_amdgcn_wmma_f32_16x16x4_f32(false, a, false, b, (short)0, c,
                                               false, false);
}

#define V8F_ZERO {0.f, 0.f, 0.f, 0.f, 0.f, 0.f, 0.f, 0.f}

// ---------------------------------------------------------------------------
// K0: Phi[b][k][h] = h_prev / delta_h for k < 64 (k-major layout: Phi^T)
// ---------------------------------------------------------------------------
__global__ void __launch_bounds__(256)
copy_base_kernel(const float* __restrict__ h_prev,
                 const float* __restrict__ delta_h,
                 float* __restrict__ phi) {
  int idx = blockIdx.x * 256 + threadIdx.x;   // BSZ*64*H elements
  int h = idx & (H_DIM - 1);
  int r = idx >> 8;                           // b*64 + k
  int k = r & 63;
  int b = r >> 6;
  float v = (k < D_DIM)
                ? h_prev[((size_t)(b * D_DIM + k)) * H_DIM + h]
                : delta_h[((size_t)(b * D_DIM + (k - D_DIM))) * H_DIM + h];
  phi[((size_t)b * KDIM + k) * H_DIM + h] = v;
}

// ---------------------------------------------------------------------------
// K1: psi GEMM  psi[p][o] = tanh( sum_h pair[p][h]*W[o][h] + bias[o] ) * w[p]
//     written directly into Phi at k = 64 + p.  f32 WMMA 16x16x4.
//     LDS holds k-PAIRED v2f elements so each fragment is one ds_load_b64.
//     Block: 128 threads (4 waves), each wave owns 64 output columns.
// ---------------------------------------------------------------------------
__global__ void __launch_bounds__(128)
psi_kernel(const float* __restrict__ z, const float* __restrict__ a_t,
           const float* __restrict__ W, const float* __restrict__ bias,
           float* __restrict__ phi) {
  // pairS2[m][kp] = {pair[m][2kp], pair[m][2kp+1]}   kp in [0,128)
  __shared__ v2f pairS2[16][130];
  // wtS2[n][kp]  = {W[n][kc+2kp], W[n][kc+2kp+1]}    kp in [0,16)
  __shared__ v2f wtS2[256][17];

  const int b    = blockIdx.y;
  const int p0   = blockIdx.x * 16;
  const int tid  = threadIdx.x;
  const int lane = tid & 31;
  const int wv   = tid >> 5;         // 0..3
  const int half = lane >> 4;
  const int m16  = lane & 15;

  // Build pair tile as v2f: pair[m][h] = z[b][i][h] * z[b][j][h]
  for (int e = tid; e < 16 * 128; e += 128) {
    int m = e >> 7, kp = e & 127;
    int p = p0 + m;
    int i = p / 31, jj = p % 31;
    int j = jj < i ? jj : jj + 1;
    v2f zi = *(const v2f*)(z + ((size_t)(b * D_DIM + i)) * H_DIM + 2 * kp);
    v2f zj = *(const v2f*)(z + ((size_t)(b * D_DIM + j)) * H_DIM + 2 * kp);
    v2f pr; pr.x = zi.x * zj.x; pr.y = zi.y * zj.y;
    pairS2[m][kp] = pr;
  }

  v8f acc0 = V8F_ZERO, acc1 = V8F_ZERO, acc2 = V8F_ZERO, acc3 = V8F_ZERO;
  const int nc = wv * 64 + m16;

  for (int kc = 0; kc < H_DIM; kc += 32) {
    __syncthreads();
    // Stage W^T chunk as v2f pairs; contiguous 8B global reads along W rows
    for (int e = tid; e < 256 * 16; e += 128) {
      int kp = e & 15, n = e >> 4;
      wtS2[n][kp] = *(const v2f*)(W + (size_t)n * H_DIM + kc + 2 * kp);
    }
    __syncthreads();
#pragma unroll
    for (int ks = 0; ks < 32; ks += 4) {
      const int kpA = (kc + ks) / 2 + half;   // K halves split across lanes
      const int kpW = ks / 2 + half;
      v2f a  = pairS2[m16][kpA];
      v2f b0 = wtS2[nc][kpW];
      v2f b1 = wtS2[nc + 16][kpW];
      v2f b2 = wtS2[nc + 32][kpW];
      v2f b3 = wtS2[nc + 48][kpW];
      acc0 = wmma_f32_k4(a, b0, acc0);
      acc1 = wmma_f32_k4(a, b1, acc1);
      acc2 = wmma_f32_k4(a, b2, acc2);
      acc3 = wmma_f32_k4(a, b3, acc3);
    }
  }

  // Epilogue: tanh(+bias) * a_t[i][j]; C/D layout rows r / r+8 per lane half
  v8f accs[4] = {acc0, acc1, acc2, acc3};
#pragma unroll
  for (int r = 0; r < 8; ++r) {
    const int m = r + 8 * half;
    const int p = p0 + m;
    const int i = p / 31, jj = p % 31;
    const int j = jj < i ? jj : jj + 1;
    const float wp = a_t[((size_t)b * D_DIM + i) * D_DIM + j];
    const size_t orow = ((size_t)b * KDIM + 2 * D_DIM + p) * H_DIM;
#pragma unroll
    for (int f = 0; f < 4; ++f) {
      const int ncol = wv * 64 + f * 16 + m16;
      phi[orow + ncol] = tanhf(accs[f][r] + bias[ncol]) * wp;
    }
  }
}

// ---------------------------------------------------------------------------
// K2: G = Phi * Phi^T  (HxH Gram over K=1056), f32 WMMA, paired-k LDS.
//     Block: 256 threads (8 waves) -> one 64x64 tile of G.
// ---------------------------------------------------------------------------
__global__ void __launch_bounds__(256)
gram_kernel(const float* __restrict__ phi, float* __restrict__ G) {
  // As2[m][kp] = {Phi^T[kc+2kp][th1+m], Phi^T[kc+2kp+1][th1+m]}
  __shared__ v2f As2[64][17];
  __shared__ v2f Bs2[64][17];

  const int b    = blockIdx.y;
  const int th1  = (blockIdx.x >> 2) * 64;
  const int th2  = (blockIdx.x & 3) * 64;
  const int tid  = threadIdx.x;
  const int lane = tid & 31;
  const int wv   = tid >> 5;         // 0..7
  const int half = lane >> 4;
  const int m16  = lane & 15;
  const int sm   = wv & 3;           // row sub-tile
  const int sn   = wv >> 2;          // col sub-tile (and sn+2)

  v8f acc0 = V8F_ZERO, acc1 = V8F_ZERO;
  const float* base = phi + (size_t)b * KDIM * H_DIM;

  for (int kc = 0; kc < KDIM; kc += 32) {
    __syncthreads();
    for (int e = tid; e < 64 * 16; e += 256) {
      int mm = e & 63, kp = e >> 6;
      size_t r0 = (size_t)(kc + 2 * kp) * H_DIM;
      v2f va; va.x = base[r0 + th1 + mm]; va.y = base[r0 + H_DIM + th1 + mm];
      v2f vb; vb.x = base[r0 + th2 + mm]; vb.y = base[r0 + H_DIM + th2 + mm];
      As2[mm][kp] = va;
      Bs2[mm][kp] = vb;
    }
    __syncthreads();
    if (kc + 32 < KDIM && tid < 32)
      __builtin_prefetch(base + (size_t)(kc + 32 + tid) * H_DIM + th1, 0, 1);
#pragma unroll
    for (int ks = 0; ks < 32; ks += 4) {
      const int kp = ks / 2 + half;
      v2f a  = As2[sm * 16 + m16][kp];
      v2f b0 = Bs2[sn * 16 + m16][kp];
      v2f b1 = Bs2[(sn + 2) * 16 + m16][kp];
      acc0 = wmma_f32_k4(a, b0, acc0);
      acc1 = wmma_f32_k4(a, b1, acc1);
    }
  }

#pragma unroll
  for (int r = 0; r < 8; ++r) {
    const int m   = r + 8 * half;
    const int row = th1 + sm * 16 + m;
    G[((size_t)b * H_DIM + row) * H_DIM + th2 + sn * 16 + m16]       = acc0[r];
    G[((size_t)b * H_DIM + row) * H_DIM + th2 + (sn + 2) * 16 + m16] = acc1[r];
  }
}

// ---------------------------------------------------------------------------
// K3: per-batch Cholesky of (G + lambda I), solve (G+lI) alpha = h_t,
//     h_hat = h_t - lambda*alpha  (dual/Woodbury identity).
// ---------------------------------------------------------------------------
__global__ void __launch_bounds__(256)
solve_kernel(const float* __restrict__ h_t, float* __restrict__ G,
             float* __restrict__ alpha, float* __restrict__ h_hat) {
  __shared__ float colk[H_DIM];
  __shared__ float yv[H_DIM];
  const int b = blockIdx.x;
  const int tid = threadIdx.x;
  float* A = G + (size_t)b * H_DIM * H_DIM;

  A[tid * H_DIM + tid] += RIDGE;
  __syncthreads();

  // In-place Cholesky (lower), column broadcast through LDS
  for (int k = 0; k < H_DIM; ++k) {
    if (tid == k) A[k * H_DIM + k] = sqrtf(A[k * H_DIM + k]);
    __syncthreads();
    const float lkk = A[k * H_DIM + k];
    float lik = 0.f;
    if (tid > k) { lik = A[tid * H_DIM + k] / lkk; A[tid * H_DIM + k] = lik; }
    colk[tid] = lik;
    __syncthreads();
    if (tid > k) {
      for (int j = k + 1; j <= tid; ++j)
        A[tid * H_DIM + j] -= lik * colk[j];
    }
    __syncthreads();
  }

  // Forward solve L y = h_t
  yv[tid] = h_t[(size_t)b * H_DIM + tid];
  __syncthreads();
  for (int k = 0; k < H_DIM; ++k) {
    if (tid == k) yv[k] /= A[k * H_DIM + k];
    __syncthreads();
    if (tid > k) yv[tid] -= A[tid * H_DIM + k] * yv[k];
    __syncthreads();
  }
  // Back solve L^T x = y  (in place in yv)
  for (int k = H_DIM - 1; k >= 0; --k) {
    if (tid == k) yv[k] /= A[k * H_DIM + k];
    __syncthreads();
    if (tid < k) yv[tid] -= A[k * H_DIM + tid] * yv[k];
    __syncthreads();
  }

  alpha[(size_t)b * H_DIM + tid] = yv[tid];
  h_hat[(size_t)b * H_DIM + tid] =
      h_t[(size_t)b * H_DIM + tid] - RIDGE * yv[tid];
}

// ---------------------------------------------------------------------------
// K4: theta[k] = sum_h Phi[b][k][h] * alpha[b][h]; route to alpha/beta/gamma.
//     One wave32 per dot, shuffle reduction.
// ---------------------------------------------------------------------------
__global__ void __launch_bounds__(256)
theta_kernel(const float* __restrict__ phi, const float* __restrict__ alpha,
             float* __restrict__ out_alpha, float* __restrict__ out_beta,
             float* __restrict__ out_gamma) {
  const int b    = blockIdx.y;
  const int tid  = threadIdx.x;
  const int lane = tid & 31;
  const int wv   = tid >> 5;
  const int k    = blockIdx.x * 8 + wv;   // < KDIM

  const float* row = phi + ((size_t)b * KDIM + k) * H_DIM;
  const float* al  = alpha + (size_t)b * H_DIM;
  float s = 0.f;
#pragma unroll
  for (int u = 0; u < 8; ++u) s += row[lane + 32 * u] * al[lane + 32 * u];
  for (int off = 16; off > 0; off >>= 1) s += __shfl_xor(s, off, 32);

  if (lane == 0) {
    if (k < D_DIM) {
      out_alpha[b * D_DIM + k] = s;
    } else if (k < 2 * D_DIM) {
      out_beta[b * D_DIM + (k - D_DIM)] = s;
    } else {
      const int p = k - 2 * D_DIM;
      const int i = p / 31, jj = p % 31;
      const int j = jj < i ? jj : jj + 1;
      out_gamma[((size_t)b * D_DIM + i) * D_DIM + j] = s;
    }
  }
  // d_out is poisoned: write gamma diagonal zeros explicitly
  if (blockIdx.x == 0 && tid < D_DIM)
    out_gamma[(size_t)b * D_DIM * D_DIM + tid * (D_DIM + 1)] = 0.f;
}

// ---------------------------------------------------------------------------
extern "C" void kernel_launch(void* const* d_in, const int* in_sizes, int n_in,
                              void* d_out, int out_size, void* d_ws,
                              size_t ws_size, hipStream_t stream) {
  (void)in_sizes; (void)n_in; (void)out_size; (void)ws_size;
  const float* h_prev  = (const float*)d_in[0];
  const float* delta_h = (const float*)d_in[1];
  const float* z       = (const float*)d_in[2];
  const float* a_t     = (const float*)d_in[3];
  const float* h_t     = (const float*)d_in[4];
  const float* W       = (const float*)d_in[5];
  const float* bias    = (const float*)d_in[6];

  float* out       = (float*)d_out;           // [h_hat | alpha | beta | gamma]
  float* h_hat     = out;                      // 64*256
  float* out_alpha = out + BSZ * H_DIM;        // 64*32
  float* out_beta  = out_alpha + BSZ * D_DIM;  // 64*32
  float* out_gamma = out_beta + BSZ * D_DIM;   // 64*32*32

  char*  ws    = (char*)d_ws;
  float* phi   = (float*)ws;                                   // 64*1056*256 f32
  size_t phiB  = (size_t)BSZ * KDIM * H_DIM * sizeof(float);
  float* G     = (float*)(ws + phiB);                          // 64*256*256 f32
  size_t gB    = (size_t)BSZ * H_DIM * H_DIM * sizeof(float);
  float* alpha = (float*)(ws + phiB + gB);                     // 64*256 f32

  copy_base_kernel<<<dim3((BSZ * 64 * H_DIM) / 256), 256, 0, stream>>>(
      h_prev, delta_h, phi);
  psi_kernel<<<dim3(NPAIR / 16, BSZ), 128, 0, stream>>>(z, a_t, W, bias, phi);
  gram_kernel<<<dim3(16, BSZ), 256, 0, stream>>>(phi, G);
  solve_kernel<<<dim3(BSZ), 256, 0, stream>>>(h_t, G, alpha, h_hat);
  theta_kernel<<<dim3(KDIM / 8, BSZ), 256, 0, stream>>>(phi, alpha, out_alpha,
                                                        out_beta, out_gamma);
}